// GraphConvolution_44624710205613
// MI455X (gfx1250) — compile-verified
//
#include <hip/hip_runtime.h>
#include <math.h>

typedef __attribute__((ext_vector_type(2))) float v2f;
typedef __attribute__((ext_vector_type(8))) float v8f;

#define NIN  256
#define NOUT 256

// ---------------------------------------------------------------------------
// Zero d_out (it doubles as the f32 atomic accumulation buffer).
// ---------------------------------------------------------------------------
__global__ __launch_bounds__(256) void zero_f4(float4* __restrict__ p, int n4) {
    int i = blockIdx.x * blockDim.x + threadIdx.x;
    if (i < n4) p[i] = make_float4(0.f, 0.f, 0.f, 0.f);
}

// ---------------------------------------------------------------------------
// h = x @ W^T + b via V_WMMA_F32_16X16X4_F32.
// One wave computes one 16x16 tile of h; K-loop steps 4.
//   A (16x4 f32):  lane m=lane&15, kh=lane>>4, VGPR j holds x[m][k + 2*kh + j]
//   B (4x16 f32):  mirrored on W^T -> contiguous float2 from W[n][.]
//   C/D (16x16 f32): VGPR v holds h[row0 + v + 8*kh][col0 + (lane&15)]
// ---------------------------------------------------------------------------
__global__ __launch_bounds__(256) void gemm_wmma_f32(
    const float* __restrict__ x, const float* __restrict__ W,
    const float* __restrict__ bias, float* __restrict__ h, int row_tiles)
{
    const int lane = threadIdx.x & 31;
    const int wave = (blockIdx.x << 3) | (threadIdx.x >> 5);
    if (wave >= row_tiles * (NOUT / 16)) return;   // wave-uniform guard

    const int row0 = (wave >> 4) << 4;
    const int col0 = (wave & 15) << 4;
    const int mn   = lane & 15;    // A row / B col
    const int kh   = lane >> 4;    // K half select

    const float* xrow = x + (size_t)(row0 + mn) * NIN + (kh << 1);
    const float* wrow = W + (size_t)(col0 + mn) * NIN + (kh << 1);

    const float bv = bias[col0 + mn];
    v8f c = { bv, bv, bv, bv, bv, bv, bv, bv };

#pragma unroll 16
    for (int k = 0; k < NIN; k += 4) {
        v2f a  = *(const v2f*)(xrow + k);
        v2f bb = *(const v2f*)(wrow + k);
        // (neg_a, A, neg_b, B, c_mod, C, reuse_a, reuse_b)
        c = __builtin_amdgcn_wmma_f32_16x16x4_f32(
                false, a, false, bb, (short)0, c, false, false);
    }

    float* hp = h + ((size_t)(row0 + kh * 8)) * NOUT + col0 + mn;
#pragma unroll
    for (int v = 0; v < 8; ++v)
        hp[(size_t)v * NOUT] = c[v];
}

// ---------------------------------------------------------------------------
// Edge scatter: one wave per edge (grid-stride). Lane i handles columns
// i + 32*j (j=0..7) so every load and every global_atomic_add_f32 is a
// contiguous 128B segment across the wave. h (102 MB) + out (102 MB) mostly
// live in the 192 MB L2, so this phase runs at L2 (not HBM) speed.
// ---------------------------------------------------------------------------
__global__ __launch_bounds__(256) void scatter_edges(
    const float* __restrict__ h, const int* __restrict__ erow,
    const int* __restrict__ ecol, const float* __restrict__ eval,
    float* __restrict__ out, int E)
{
    const int lane = threadIdx.x & 31;
    const int wid  = __builtin_amdgcn_readfirstlane(
        blockIdx.x * (blockDim.x >> 5) + (threadIdx.x >> 5));
    const int nw   = gridDim.x * (blockDim.x >> 5);

    for (int e = wid; e < E; e += nw) {
        const int   c = ecol[e];
        const int   r = erow[e];
        const float v = eval[e];

        // Prefetch next edge's h row into cache (global_prefetch_b8).
        const int en = e + nw;
        if (en < E) {
            const int cn = ecol[en];
            __builtin_prefetch(h + (size_t)cn * NOUT + lane * 8, 0, 3);
        }

        const float* hp = h   + (size_t)c * NOUT + lane;
        float*       op = out + (size_t)r * NOUT + lane;
#pragma unroll
        for (int j = 0; j < 8; ++j) {
            atomicAdd(op + j * 32, v * hp[j * 32]);
        }
    }
}

// ---------------------------------------------------------------------------
// In-place ELU epilogue: out = x > 0 ? x : exp(x) - 1
// ---------------------------------------------------------------------------
__global__ __launch_bounds__(256) void elu_inplace(float4* __restrict__ p, int n4) {
    int i = blockIdx.x * blockDim.x + threadIdx.x;
    if (i >= n4) return;
    float4 a = p[i];
    a.x = a.x > 0.f ? a.x : __expf(a.x) - 1.f;
    a.y = a.y > 0.f ? a.y : __expf(a.y) - 1.f;
    a.z = a.z > 0.f ? a.z : __expf(a.z) - 1.f;
    a.w = a.w > 0.f ? a.w : __expf(a.w) - 1.f;
    p[i] = a;
}

// ---------------------------------------------------------------------------
extern "C" void kernel_launch(void* const* d_in, const int* in_sizes, int n_in,
                              void* d_out, int out_size, void* d_ws, size_t ws_size,
                              hipStream_t stream) {
    const float* x    = (const float*)d_in[0];
    const int*   erow = (const int*)  d_in[1];
    const int*   ecol = (const int*)  d_in[2];
    const float* eval = (const float*)d_in[3];
    const float* W    = (const float*)d_in[4];
    const float* b    = (const float*)d_in[5];
    float*       out  = (float*)d_out;
    float*       h    = (float*)d_ws;        // N*256 f32 = 102.4 MB scratch

    const int N = in_sizes[0] / NIN;         // 100000
    const int E = in_sizes[1];               // 3200000
    const int n4 = (N * NOUT) / 4;

    // 1) zero the accumulation buffer (d_out)
    zero_f4<<<(n4 + 255) / 256, 256, 0, stream>>>((float4*)out, n4);

    // 2) dense projection h = x W^T + b  (WMMA f32 16x16x4)
    const int row_tiles   = N / 16;                 // 6250
    const int total_tiles = row_tiles * (NOUT / 16); // 100000 waves
    gemm_wmma_f32<<<(total_tiles + 7) / 8, 256, 0, stream>>>(x, W, b, h, row_tiles);

    // 3) edge gather + atomic scatter into d_out
    scatter_edges<<<8192, 256, 0, stream>>>(h, erow, ecol, eval, out, E);

    // 4) ELU in place
    elu_inplace<<<(n4 + 255) / 256, 256, 0, stream>>>((float4*)out, n4);
}